// LM2MemoryModule_Explan_39135742002009
// MI455X (gfx1250) — compile-verified
//
#include <hip/hip_runtime.h>
#include <hip/hip_bf16.h>

typedef __attribute__((ext_vector_type(2))) float v2f;
typedef __attribute__((ext_vector_type(8))) float v8f;

#define D 256
#define NSLOT 8
#define BATCH 2
#define SEQ 64

// workspace layout (in floats)
#define OFF_GK 0
#define OFF_GV (NSLOT * D * D)                    // 524288
#define OFF_RK (2 * NSLOT * D * D)                // 1048576
#define OFF_RV (2 * NSLOT * D * D + D * D)        // 1114112
#define OFF_F  (2 * NSLOT * D * D + 2 * D * D)    // 1179648
#define OFF_A  (OFF_F + BATCH * D)                // 1180160

// ---------------------------------------------------------------------------
// R_P[r,j] = sum_c W_P[(r*D+c)*D + j]   for P in {K, V}
// ---------------------------------------------------------------------------
__global__ void rowsum_kernel(const float* __restrict__ WK,
                              const float* __restrict__ WV,
                              float* __restrict__ ws) {
    const int r = blockIdx.x;
    const int j = threadIdx.x;
    const float* W = blockIdx.y ? WV : WK;
    float* R = ws + (blockIdx.y ? OFF_RV : OFF_RK);
    const float* p = W + (size_t)r * D * D + j;
    float s = 0.f;
#pragma unroll 8
    for (int c = 0; c < D; ++c) s += p[(size_t)c * D];
    R[r * D + j] = s;
}

// ---------------------------------------------------------------------------
// G_P[n,r,j] = sum_c M0[n,r,c] * W_P[(r*D+c)*D + j]
// Exact fp32 via V_WMMA_F32_16X16X4_F32. One wave per (r, j-tile, {K,V}).
// Padded rows (m>=8) are handled by loading an aliased valid row and
// multiplying the fragment by a 0/1 lane mask -- no divergent loads, so the
// k-loop runs with EXEC all-ones and no exec save/restore branches.
// Two independent accumulators over the two K halves hide the wmma->wmma
// dependency behind the operand loads.
// ---------------------------------------------------------------------------
__global__ void __launch_bounds__(32)
gmat_wmma_kernel(const float* __restrict__ M0,
                 const float* __restrict__ WK,
                 const float* __restrict__ WV,
                 float* __restrict__ ws) {
    const int r  = blockIdx.x;        // 0..255
    const int j0 = blockIdx.y * 16;   // j tile base
    const float* W = blockIdx.z ? WV : WK;
    float* G = ws + (blockIdx.z ? OFF_GV : OFF_GK);

    const int lane = threadIdx.x;         // 0..31
    const int m    = lane & 15;           // A row (padded to 16)
    const int kh   = (lane >> 4) * 2;     // K-pair offset (0 or 2)
    const int jc   = lane & 15;           // B column
    const float msk = (m < NSLOT) ? 1.f : 0.f;

    // A[m,k] = M0[m*D*D + r*D + k]; aliased row (m&7) for padded lanes, masked.
    const float* Ap = M0 + (size_t)(m & 7) * D * D + (size_t)r * D + kh;
    // B[k,j] = W[(r*D+k)*D + j]; pre-offset by kh rows and the column.
    const float* Wb = W + (size_t)r * D * D + (size_t)kh * D + (j0 + jc);

    v8f acc0 = {}, acc1 = {};
#pragma unroll 4
    for (int k0 = 0; k0 < D / 2; k0 += 4) {
        const int k1 = k0 + D / 2;

        v2f a0 = *(const v2f*)(Ap + k0);      // one b64 load: A[m,k0+kh..+1]
        a0 *= msk;
        v2f b0;
        b0.x = Wb[(size_t)(k0 + 0) * D];
        b0.y = Wb[(size_t)(k0 + 1) * D];
        acc0 = __builtin_amdgcn_wmma_f32_16x16x4_f32(
            false, a0, false, b0, (short)0, acc0, false, false);

        v2f a1 = *(const v2f*)(Ap + k1);
        a1 *= msk;
        v2f b1;
        b1.x = Wb[(size_t)(k1 + 0) * D];
        b1.y = Wb[(size_t)(k1 + 1) * D];
        acc1 = __builtin_amdgcn_wmma_f32_16x16x4_f32(
            false, a1, false, b1, (short)0, acc1, false, false);
    }
    const v8f acc = acc0 + acc1;

    // C/D layout: lanes 0-15 hold N=lane, M = vgpr index 0..7 (rows we need).
    if (lane < 16) {
#pragma unroll
        for (int mm = 0; mm < NSLOT; ++mm)
            G[(size_t)mm * D * D + (size_t)r * D + (j0 + lane)] = acc[mm];
    }
}

// ---------------------------------------------------------------------------
// Sequential recurrence: one workgroup per batch, 256 threads (thread = col j)
// ---------------------------------------------------------------------------
__global__ void __launch_bounds__(256)
lm2_scan_kernel(const float* __restrict__ E_t,
                const float* __restrict__ W_Q, const float* __restrict__ b_Q,
                const float* __restrict__ b_K, const float* __restrict__ b_V,
                const float* __restrict__ W_out, const float* __restrict__ b_out,
                const float* __restrict__ W_fg, const float* __restrict__ b_fg,
                const float* __restrict__ W_in, const float* __restrict__ b_in,
                const float* __restrict__ GK, const float* __restrict__ GV,
                const float* __restrict__ RK, const float* __restrict__ RV,
                float* __restrict__ Fout, float* __restrict__ Aout,
                float* __restrict__ E_out) {
    __shared__ float e[D], q[D], Emem[D], Fv[D], Av[D];
    __shared__ float Kv[NSLOT][D], Vv[NSLOT][D];
    __shared__ float s_sc[NSLOT], p_sc[NSLOT];

    const int b = blockIdx.x;
    const int j = threadIdx.x;

    Fv[j] = 1.f;
    Av[j] = 0.f;
    e[j]  = E_t[(size_t)b * SEQ * D + j];
    __syncthreads();

    for (int t = 0; t < SEQ; ++t) {
        // ---- phase 1: Q and A-dependent base terms (thread owns column j)
        float qj = b_Q[j], bk = b_K[j], bv = b_V[j];
#pragma unroll 4
        for (int r = 0; r < D; ++r) {
            const float er = e[r], ar = Av[r];
            qj += er * W_Q[r * D + j];
            bk += ar * RK[r * D + j];
            bv += ar * RV[r * D + j];
        }
        q[j] = qj;

        // ---- phase 2: per-slot K, V  (K[n,j] = base + F·G_K[n,:,j])
        for (int n = 0; n < NSLOT; ++n) {
            float kn = bk, vn = bv;
            const float* gk = GK + (size_t)n * D * D + j;
            const float* gv = GV + (size_t)n * D * D + j;
#pragma unroll 4
            for (int r = 0; r < D; ++r) {
                const float fr = Fv[r];
                kn += fr * gk[(size_t)r * D];
                vn += fr * gv[(size_t)r * D];
            }
            Kv[n][j] = kn;
            Vv[n][j] = vn;
        }
        __syncthreads();

        // ---- phase 3: scores, one wave32 per slot
        {
            const int w = j >> 5, lane = j & 31;
            float part = 0.f;
            for (int jj = lane; jj < D; jj += 32) part += q[jj] * Kv[w][jj];
            for (int off = 16; off > 0; off >>= 1)
                part += __shfl_down(part, off, 32);
            if (lane == 0) s_sc[w] = part * 0.0625f;   // 1/sqrt(256)
        }
        __syncthreads();

        // ---- phase 4: softmax over the 8 slots
        if (j == 0) {
            float mx = s_sc[0];
            for (int n = 1; n < NSLOT; ++n) mx = fmaxf(mx, s_sc[n]);
            float sum = 0.f;
            for (int n = 0; n < NSLOT; ++n) { float ex = expf(s_sc[n] - mx); p_sc[n] = ex; sum += ex; }
            const float inv = 1.f / sum;
            for (int n = 0; n < NSLOT; ++n) p_sc[n] *= inv;
        }
        __syncthreads();

        // ---- phase 5: E_mem = sum_n p[n] * V[n]
        float em = 0.f;
#pragma unroll
        for (int n = 0; n < NSLOT; ++n) em += p_sc[n] * Vv[n][j];
        Emem[j] = em;
        __syncthreads();

        // ---- phase 6: gates + output token
        float xo = b_out[j], xf = b_fg[j], xi = b_in[j];
#pragma unroll 4
        for (int r = 0; r < D; ++r) {
            const float mr = Emem[r], er = e[r];
            xo += mr * W_out[r * D + j];
            xf += mr * W_fg[r * D + j];
            xi += er * W_in[r * D + j];
        }
        const float go = 1.f / (1.f + expf(-xo));
        const float gf = 1.f / (1.f + expf(-xf));
        const float gi = 1.f / (1.f + expf(-xi));
        E_out[(size_t)b * SEQ * D + (size_t)t * D + j] = e[j] + go * em;
        const float nF = gf * Fv[j];
        const float nA = gi * tanhf(em) + gf * Av[j];
        __syncthreads();                 // all reads of e/Fv/Av/Emem done

        // ---- phase 7: advance state
        Fv[j] = nF;
        Av[j] = nA;
        if (t + 1 < SEQ) e[j] = E_t[(size_t)b * SEQ * D + (size_t)(t + 1) * D + j];
        __syncthreads();
    }

    Fout[b * D + j] = Fv[j];
    Aout[b * D + j] = Av[j];
}

// ---------------------------------------------------------------------------
// M_out[b,n,r,c] = F[b,r] * M0[n,r,c] + A[b,r]
// ---------------------------------------------------------------------------
__global__ void final_m_kernel(const float* __restrict__ M0,
                               const float* __restrict__ ws,
                               float* __restrict__ Mout) {
    const int idx = blockIdx.x * blockDim.x + threadIdx.x;  // < 2*8*256*256
    const int c = idx & 255;
    const int r = (idx >> 8) & 255;
    const int n = (idx >> 16) & 7;
    const int b = idx >> 19;
    Mout[idx] = ws[OFF_F + b * D + r] * M0[(size_t)n * D * D + r * D + c]
              + ws[OFF_A + b * D + r];
}

extern "C" void kernel_launch(void* const* d_in, const int* in_sizes, int n_in,
                              void* d_out, int out_size, void* d_ws, size_t ws_size,
                              hipStream_t stream) {
    (void)in_sizes; (void)n_in; (void)out_size; (void)ws_size;
    const float* E_t      = (const float*)d_in[0];
    const float* memory   = (const float*)d_in[1];
    const float* W_Q      = (const float*)d_in[2];
    const float* b_Q      = (const float*)d_in[3];
    const float* W_K      = (const float*)d_in[4];
    const float* b_K      = (const float*)d_in[5];
    const float* W_V      = (const float*)d_in[6];
    const float* b_V      = (const float*)d_in[7];
    const float* W_out    = (const float*)d_in[8];
    const float* b_out    = (const float*)d_in[9];
    const float* W_fg     = (const float*)d_in[10];
    const float* b_fg     = (const float*)d_in[11];
    const float* W_in     = (const float*)d_in[12];
    const float* b_in     = (const float*)d_in[13];
    float* out = (float*)d_out;
    float* ws  = (float*)d_ws;

    rowsum_kernel<<<dim3(256, 2), 256, 0, stream>>>(W_K, W_V, ws);
    gmat_wmma_kernel<<<dim3(256, 16, 2), 32, 0, stream>>>(memory, W_K, W_V, ws);
    lm2_scan_kernel<<<BATCH, D, 0, stream>>>(
        E_t, W_Q, b_Q, b_K, b_V, W_out, b_out, W_fg, b_fg, W_in, b_in,
        ws + OFF_GK, ws + OFF_GV, ws + OFF_RK, ws + OFF_RV,
        ws + OFF_F, ws + OFF_A,
        out /* E_out occupies first B*SEQ*D floats */);
    final_m_kernel<<<(BATCH * NSLOT * D * D) / 256, 256, 0, stream>>>(
        memory, ws, out + BATCH * SEQ * D);
}